// InvariantGeometricFeatures_12343736009198
// MI455X (gfx1250) — compile-verified
//
#include <hip/hip_runtime.h>
#include <hip/hip_bf16.h>

// Problem constants (reference is fixed-shape: x [4, 3, 8192])
#define NB    4
#define NPTS  8192
#define KNN   20
#define TILES (NPTS / 16)   // 512 column chunks / row tiles

typedef __attribute__((ext_vector_type(2))) float v2f;
typedef __attribute__((ext_vector_type(8))) float v8f;

// One wave per (batch, 16-row tile). WMMA 16x16x4 f32 computes, per tile,
//   acc[col][row] = sq_col - 2 * col . row        (sq and -2 folded into K=4)
// so d2 = acc + sq_row. Each lane owns 8 candidates of one fixed row per
// chunk; per-row top-20-smallest is kept in that lane's registers.
__global__ __launch_bounds__(32) void knn_rowstats_kernel(
    const float* __restrict__ x, float4* __restrict__ rowstats) {
  const int lane = threadIdx.x;        // 0..31
  const int half = lane >> 4;          // 0: K={0,1} slots, 1: K={2,3} slots
  const int l16  = lane & 15;
  const int tile = blockIdx.x % TILES;
  const int b    = blockIdx.x / TILES;

  const float* X0 = x + (size_t)b * 3 * NPTS;   // component planes of [B,3,N]
  const float* X1 = X0 + NPTS;
  const float* X2 = X1 + NPTS;

  // B-matrix (4 x 16): rows of this tile, fixed for the whole wave.
  // lanes 0-15: (r0, r1) of row l16 ; lanes 16-31: (r2, 1.0)  [K=3 slot = 1]
  const int row = tile * 16 + l16;
  const float r0 = X0[row], r1 = X1[row], r2 = X2[row];
  v2f bmat;
  bmat.x = half ? r2 : r0;
  bmat.y = half ? 1.0f : r1;
  const float sqr = r0 * r0 + r1 * r1 + r2 * r2;  // |row point|^2 (lane's row)

  float t[KNN];
#pragma unroll
  for (int i = 0; i < KNN; ++i) t[i] = __builtin_inff();
  float mx    = __builtin_inff();   // max of t[]
  float mxAdj = __builtin_inff();   // mx - sqr (filter threshold on acc values)

  for (int ck = 0; ck < TILES; ++ck) {
    // A-matrix (16 cols x 4) for this chunk:
    // lanes 0-15: (-2c0, -2c1) of col l16 ; lanes 16-31: (-2c2, csq)
    const int col = ck * 16 + l16;
    const float c0 = X0[col], c1 = X1[col], c2 = X2[col];
    const float csq = c0 * c0 + c1 * c1 + c2 * c2;
    v2f amat;
    amat.x = half ? (-2.0f * c2) : (-2.0f * c0);
    amat.y = half ? csq          : (-2.0f * c1);

    v8f acc = {0.f, 0.f, 0.f, 0.f, 0.f, 0.f, 0.f, 0.f};
    // acc[m=col][n=row] = csq_col - 2 * colPt . rowPt
    acc = __builtin_amdgcn_wmma_f32_16x16x4_f32(
        false, amat, false, bmat, (short)0, acc, false, false);

    // C/D layout: lane (l<16): vgpr v -> (col v, row l);
    //             lane (l>=16): vgpr v -> (col v+8, row l-16).
#pragma unroll
    for (int v = 0; v < 8; ++v) {
      if (acc[v] < mxAdj) {            // i.e. d2 = acc[v] + sqr < mx
        const float d2 = acc[v] + sqr;
        // replace current max element (first match), no dynamic indexing
        bool done = false;
#pragma unroll
        for (int i = 0; i < KNN; ++i) {
          const bool hit = (!done) && (t[i] == mx);
          t[i] = hit ? d2 : t[i];
          done = done || hit;
        }
        float m = t[0];
#pragma unroll
        for (int i = 1; i < KNN; ++i) m = fmaxf(m, t[i]);
        mx    = m;
        mxAdj = mx - sqr;
      }
    }
  }

  // Merge lane l and lane l+16 (the two column-halves of row l): the union of
  // their 40 candidates provably contains the row's true 20 smallest.
  __shared__ float mrg[32 * KNN];
#pragma unroll
  for (int i = 0; i < KNN; ++i) mrg[lane * KNN + i] = t[i];
  __syncthreads();

  if (lane < 16) {
    float sum = 0.f, sumsq = 0.f, dmin = 0.f, dmax = 0.f;
    for (int k = 0; k < KNN; ++k) {
      float mv = __builtin_inff();
      int mj = lane * KNN;
#pragma unroll 4
      for (int j = 0; j < 2 * KNN; ++j) {
        const int idx = (j < KNN) ? (lane * KNN + j) : ((lane + 16) * KNN + (j - KNN));
        const float vv = mrg[idx];
        if (vv < mv) { mv = vv; mj = idx; }
      }
      mrg[mj] = __builtin_inff();          // extract-min
      const float d2c = fmaxf(mv, 0.0f);   // clamp like reference
      const float d   = sqrtf(d2c);
      sum   += d;
      sumsq += d2c;                        // sum of dist^2 == sum of clamped d2
      if (k == 0)       dmin = d;          // nearest (self) distance
      if (k == KNN - 1) dmax = d;          // 20th smallest distance
    }
    rowstats[(size_t)b * NPTS + row] = make_float4(sum, sumsq, dmin, dmax);
  }
}

// Reduce rowstats -> global mean/var of knn values; fold conv+BN+gamma/beta
// into per-channel affine (alpha_c, delta_c). conv bias cancels in (feat-mean).
__global__ __launch_bounds__(256) void bn_coef_kernel(
    const float4* __restrict__ rowstats, const float* __restrict__ w,
    const float* __restrict__ bias, const float* __restrict__ gamma,
    const float* __restrict__ beta, float* __restrict__ coef) {
  __shared__ float sA[256], sQ[256];
  const int tid = threadIdx.x;
  float a = 0.f, q = 0.f;
  for (int i = tid; i < NB * NPTS; i += 256) {
    const float4 rs = rowstats[i];
    a += rs.x;
    q += rs.y;
  }
  sA[tid] = a; sQ[tid] = q;
  __syncthreads();
  for (int s = 128; s > 0; s >>= 1) {
    if (tid < s) { sA[tid] += sA[tid + s]; sQ[tid] += sQ[tid + s]; }
    __syncthreads();
  }
  if (tid == 0) {
    const float cnt = (float)(NB * NPTS * KNN);
    const float mu  = sA[0] / cnt;
    const float var = fmaxf(sQ[0] / cnt - mu * mu, 0.0f);  // biased variance
    (void)bias;  // cancels: feat - mean_c = w_c*(d - mu)
    for (int c = 0; c < 16; ++c) {
      const float wc    = w[c];
      const float alpha = gamma[c] * wc * rsqrtf(wc * wc * var + 1e-5f);
      coef[c]      = alpha;
      coef[16 + c] = beta[c] - alpha * mu;
    }
  }
}

// out[b,c,n] = lrelu(alpha_c * (alpha_c>=0 ? dmax : dmin) + delta_c)
// (LeakyReLU is monotone increasing => max over k moves inside the affine map.)
__global__ __launch_bounds__(256) void out_kernel(
    const float4* __restrict__ rowstats, const float* __restrict__ coef,
    float* __restrict__ out) {
  const int idx = blockIdx.x * 256 + threadIdx.x;   // = ((b*16 + c)*NPTS + n)
  const int n = idx & (NPTS - 1);
  const int c = (idx >> 13) & 15;
  const int b = idx >> 17;
  const float4 rs   = rowstats[(size_t)b * NPTS + n];
  const float alpha = coef[c];
  const float delta = coef[16 + c];
  const float d = (alpha >= 0.f) ? rs.w : rs.z;
  const float y = fmaf(alpha, d, delta);
  out[idx] = (y >= 0.f) ? y : 0.2f * y;
}

extern "C" void kernel_launch(void* const* d_in, const int* in_sizes, int n_in,
                              void* d_out, int out_size, void* d_ws, size_t ws_size,
                              hipStream_t stream) {
  (void)in_sizes; (void)n_in; (void)out_size; (void)ws_size;
  const float* x      = (const float*)d_in[0];   // [4,3,8192]
  const float* conv_w = (const float*)d_in[1];   // [16]
  const float* conv_b = (const float*)d_in[2];   // [16]
  const float* gamma  = (const float*)d_in[3];   // [16]
  const float* beta   = (const float*)d_in[4];   // [16]
  float* out          = (float*)d_out;           // [4,16,8192]

  float4* rowstats = (float4*)d_ws;                                      // 32768 * 16 B
  float*  coef     = (float*)((char*)d_ws + sizeof(float4) * NB * NPTS); // 32 floats

  knn_rowstats_kernel<<<NB * TILES, 32, 0, stream>>>(x, rowstats);
  bn_coef_kernel<<<1, 256, 0, stream>>>(rowstats, conv_w, conv_b, gamma, beta, coef);
  out_kernel<<<(NB * 16 * NPTS) / 256, 256, 0, stream>>>(rowstats, coef, out);
}